// EUNNCell_82454782149321
// MI455X (gfx1250) — compile-verified
//
#include <hip/hip_runtime.h>
#include <hip/hip_bf16.h>
#include <math.h>
#include <stdint.h>

// EUNN cell for MI455X (gfx1250, wave32).
// M = composition of 128 Givens layers (banded, half-bandwidth 128), applied to
// the batch as a banded complex GEMM on v_wmma_f32_16x16x4_f32 (fp32 precision).
//
// Stage 1: per-layer rotation coefficients + cos/sin(omega) -> workspace.
// Stage 2: push H identity columns through the 128 layers (1 wave/column,
//          32 complex elems/lane in VGPRs, lane shuffles at chunk edges).
//          Writes M transposed & row-pair interleaved: MtP[j/2][i][j&1].
// Stage 3: banded complex GEMM, 2x2 tile blocking per wave (16 WMMA / K-step).
//          All 8 waves of a block share one 32-wide i-group; the Mt band is
//          double-buffer staged into LDS with global_load_async_to_lds_b128
//          (ASYNCcnt) and consumed as ds_load_b64 B-operands.

#define HDIM 1024
#define BDIM 2048
#define NSTEP 64

typedef float v2f __attribute__((ext_vector_type(2)));
typedef float v8f __attribute__((ext_vector_type(8)));

static __device__ __forceinline__ v8f wmma_f32(v2f a, v2f b, v8f c) {
  return __builtin_amdgcn_wmma_f32_16x16x4_f32(false, a, false, b, (short)0, c,
                                               false, false);
}

// ---------------------------------------------------------------------------
// Stage 1: coefficient vectors. coeff[s][0=even,1=odd][v1r,v1i,v2r,v2i][k]
// ---------------------------------------------------------------------------
__global__ void __launch_bounds__(256)
eunn_precompute(const float* __restrict__ omega,
                const float* __restrict__ eth, const float* __restrict__ oth,
                const float* __restrict__ eph, const float* __restrict__ oph,
                float* __restrict__ coeff,
                float* __restrict__ cw, float* __restrict__ sw) {
  int tid = blockIdx.x * blockDim.x + threadIdx.x;
  int s = tid >> 10;
  int k = tid & (HDIM - 1);
  if (s >= NSTEP) return;

  { // even layer: pairs (2m, 2m+1)
    int m = k >> 1;
    float th = eth[s * 512 + m], ph = eph[s * 512 + m];
    float ct = cosf(th), st = sinf(th), cp = cosf(ph), sp = sinf(ph);
    float v1r, v1i, v2r, v2i;
    if ((k & 1) == 0) { v1r = sp * ct; v1i = cp * ct; v2r = -sp * st; v2i = -cp * st; }
    else              { v1r = ct;      v1i = 0.f;     v2r = st;       v2i = 0.f;      }
    float* cb = coeff + (size_t)(s * 2 + 0) * 4 * HDIM;
    cb[0 * HDIM + k] = v1r; cb[1 * HDIM + k] = v1i;
    cb[2 * HDIM + k] = v2r; cb[3 * HDIM + k] = v2i;
  }
  { // odd layer: identity at ends, pairs (1,2),(3,4),...
    float v1r = 1.f, v1i = 0.f, v2r = 0.f, v2i = 0.f;
    if (k > 0 && k < HDIM - 1) {
      int m = (k - 1) >> 1;
      float th = oth[s * 511 + m], ph = oph[s * 511 + m];
      float ct = cosf(th), st = sinf(th), cp = cosf(ph), sp = sinf(ph);
      if (k & 1) { v1r = sp * ct; v1i = cp * ct; v2r = -sp * st; v2i = -cp * st; }
      else       { v1r = ct;      v1i = 0.f;     v2r = st;       v2i = 0.f;      }
    }
    float* cb = coeff + (size_t)(s * 2 + 1) * 4 * HDIM;
    cb[0 * HDIM + k] = v1r; cb[1 * HDIM + k] = v1i;
    cb[2 * HDIM + k] = v2r; cb[3 * HDIM + k] = v2i;
  }
  if (s == 0) { cw[k] = cosf(omega[k]); sw[k] = sinf(omega[k]); }
}

// ---------------------------------------------------------------------------
// Stage 2: MtP[j>>1][i][j&1] = (composed operator applied to e_j)[i].
// ---------------------------------------------------------------------------
__global__ void __launch_bounds__(256)
eunn_build(const float* __restrict__ coeff,
           float* __restrict__ MtrP, float* __restrict__ MtiP) {
  const int j  = blockIdx.x * (blockDim.x >> 5) + (threadIdx.x >> 5);
  const int ln = threadIdx.x & 31;
  if (j >= HDIM) return;
  const int base = ln << 5;

  float re[32], im[32], nr[32], ni[32];
#pragma unroll
  for (int t = 0; t < 32; ++t) {
    re[t] = (base + t == j) ? 1.f : 0.f;
    im[t] = 0.f;
  }

  for (int s = 0; s < NSTEP; ++s) {
    const float* ce = coeff + (size_t)(s * 2 + 0) * 4 * HDIM;
#pragma unroll
    for (int t = 0; t < 32; ++t) {
      const int k = base + t;
      const float v1r = ce[0 * HDIM + k], v1i = ce[1 * HDIM + k];
      const float v2r = ce[2 * HDIM + k], v2i = ce[3 * HDIM + k];
      const int p = t ^ 1;
      nr[t] = v1r * re[t] - v1i * im[t] + v2r * re[p] - v2i * im[p];
      ni[t] = v1i * re[t] + v1r * im[t] + v2i * re[p] + v2r * im[p];
    }
#pragma unroll
    for (int t = 0; t < 32; ++t) { re[t] = nr[t]; im[t] = ni[t]; }

    const float* co = coeff + (size_t)(s * 2 + 1) * 4 * HDIM;
    const float prR = __shfl_down(re[0], 1, 32);
    const float prI = __shfl_down(im[0], 1, 32);
    const float plR = __shfl_up(re[31], 1, 32);
    const float plI = __shfl_up(im[31], 1, 32);
#pragma unroll
    for (int t = 0; t < 32; ++t) {
      const int k = base + t;
      const float v1r = co[0 * HDIM + k], v1i = co[1 * HDIM + k];
      const float v2r = co[2 * HDIM + k], v2i = co[3 * HDIM + k];
      float pRe, pIm;
      if (t == 0)       { pRe = plR;       pIm = plI;       }
      else if (t == 31) { pRe = prR;       pIm = prI;       }
      else if (t & 1)   { pRe = re[t + 1]; pIm = im[t + 1]; }
      else              { pRe = re[t - 1]; pIm = im[t - 1]; }
      nr[t] = v1r * re[t] - v1i * im[t] + v2r * pRe - v2i * pIm;
      ni[t] = v1i * re[t] + v1r * im[t] + v2i * pRe + v2r * pIm;
    }
#pragma unroll
    for (int t = 0; t < 32; ++t) { re[t] = nr[t]; im[t] = ni[t]; }
  }

  // pair-interleaved transpose store: MtP[(j>>1)*2H + i*2 + (j&1)]
  float* rowR = MtrP + (size_t)(j >> 1) * (2 * HDIM) + (j & 1);
  float* rowI = MtiP + (size_t)(j >> 1) * (2 * HDIM) + (j & 1);
#pragma unroll
  for (int t = 0; t < 32; ++t) {
    rowR[(size_t)(base + t) * 2] = re[t];
    rowI[(size_t)(base + t) * 2] = im[t];
  }
}

// ---------------------------------------------------------------------------
// Stage 3: banded complex GEMM + omega phase.
// Block = 8 waves, all sharing one i-group (32 i). Wave computes 32i x 32b.
//   A (16x4 f32): lane -> b=b0+(ln&15), K=2*(ln>>4)+{0,1}  (one b64 load)
//   B (4x16 f32): V0->K=2*(ln>>4), V1->K+1, N=i            (one ds b64 load)
//   D (16x16):    VGPR r -> b=b0+r+8*(ln>>4), i=i0+(ln&15)
// ---------------------------------------------------------------------------
__global__ void __launch_bounds__(256)
eunn_gemm(const float* __restrict__ xr, const float* __restrict__ xi,
          const float* __restrict__ MtrP, const float* __restrict__ MtiP,
          const float* __restrict__ cw, const float* __restrict__ sw,
          float* __restrict__ out) {
  // LDS: [buf][re/im][pair-row 0..7][i 0..31][q 0..1] = 8 KB total
  __shared__ float LB[2][2][8][64];

  const int wid  = threadIdx.x >> 5;
  const int ln   = threadIdx.x & 31;
  const int lo   = ln & 15, hf = ln >> 4;

  const int ig = blockIdx.x >> 3;                  // 32 i-groups of 32
  const int bg = (blockIdx.x & 7) * 8 + wid;       // 64 b-groups of 32
  const int I0 = ig << 5;
  const int B0 = bg << 5;

  // exact band for this i-group: |i-j| <= 128, multiple of 32 -> 16-chunks exact
  const int jlo = (I0 >= 128) ? (I0 - 128) : 0;
  int jhi = I0 + 160; if (jhi > HDIM) jhi = HDIM;
  const int nchunk = (jhi - jlo) >> 4;             // 16 j-rows per chunk

  // async staging of one 16-row x 32-col chunk (pair-interleaved, 4 KB)
  const int s_reim = threadIdx.x >> 7;             // 0: re, 1: im
  const int s_rem  = threadIdx.x & 127;
  const int s_prel = s_rem >> 4;                   // pair-row 0..7
  const int s_off  = s_rem & 15;                   // 16B unit within 256B row
  const float* s_base = (s_reim ? MtiP : MtrP);
  {
    // prologue: stage chunk 0 into buffer 0
    const float* src = s_base + (size_t)((jlo >> 1) + s_prel) * (2 * HDIM)
                              + I0 * 2 + s_off * 4;
    uint32_t dst = (uint32_t)(uintptr_t)&LB[0][s_reim][s_prel][s_off * 4];
    asm volatile("global_load_async_to_lds_b128 %0, %1, off"
                 :: "v"(dst), "v"(src) : "memory");
  }
  asm volatile("s_wait_asynccnt 0x0" ::: "memory");
  __syncthreads();

  v8f aRR[2][2], aII[2][2], aRI[2][2], aIR[2][2];
#pragma unroll
  for (int it = 0; it < 2; ++it)
#pragma unroll
    for (int bt = 0; bt < 2; ++bt) {
      aRR[it][bt] = (v8f){}; aII[it][bt] = (v8f){};
      aRI[it][bt] = (v8f){}; aIR[it][bt] = (v8f){};
    }

  const float* xrp[2]; const float* xip[2];
#pragma unroll
  for (int bt = 0; bt < 2; ++bt) {
    xrp[bt] = xr + (size_t)(B0 + bt * 16 + lo) * HDIM + 2 * hf;
    xip[bt] = xi + (size_t)(B0 + bt * 16 + lo) * HDIM + 2 * hf;
  }

  for (int c = 0; c < nchunk; ++c) {
    const int buf = c & 1;
    const int jb = jlo + (c << 4);                 // global j base of chunk

    if (c + 1 < nchunk) {                          // stage next chunk (other buf)
      const float* src = s_base + (size_t)(((jb + 16) >> 1) + s_prel) * (2 * HDIM)
                                + I0 * 2 + s_off * 4;
      uint32_t dst = (uint32_t)(uintptr_t)&LB[buf ^ 1][s_reim][s_prel][s_off * 4];
      asm volatile("global_load_async_to_lds_b128 %0, %1, off"
                   :: "v"(dst), "v"(src) : "memory");
    }
    // prefetch the A streams one chunk ahead
    __builtin_prefetch(xrp[0] + jb + 16, 0, 1);
    __builtin_prefetch(xip[0] + jb + 16, 0, 1);
    __builtin_prefetch(xrp[1] + jb + 16, 0, 1);
    __builtin_prefetch(xip[1] + jb + 16, 0, 1);

#pragma unroll
    for (int kk = 0; kk < 4; ++kk) {               // 4 K-steps of K=4
      const int j0 = jb + kk * 4;
      const int prel = kk * 2 + hf;                // LDS pair-row for this lane
      v2f ar[2], ai[2];
#pragma unroll
      for (int bt = 0; bt < 2; ++bt) {
        ar[bt] = *(const v2f*)(xrp[bt] + j0);
        ai[bt] = *(const v2f*)(xip[bt] + j0);
      }
#pragma unroll
      for (int it = 0; it < 2; ++it) {
        const int irel = it * 16 + lo;
        v2f br = *(const v2f*)&LB[buf][0][prel][irel * 2];
        v2f bi = *(const v2f*)&LB[buf][1][prel][irel * 2];
#pragma unroll
        for (int bt = 0; bt < 2; ++bt) {
          aRR[it][bt] = wmma_f32(ar[bt], br, aRR[it][bt]);
          aII[it][bt] = wmma_f32(ai[bt], bi, aII[it][bt]);
          aRI[it][bt] = wmma_f32(ar[bt], bi, aRI[it][bt]);
          aIR[it][bt] = wmma_f32(ai[bt], br, aIR[it][bt]);
        }
      }
    }
    asm volatile("s_wait_asynccnt 0x0" ::: "memory");
    __syncthreads();
  }

  // epilogue: complex combine + e^{i*omega_i} phase
  float* outre = out;
  float* outim = out + (size_t)BDIM * HDIM;
#pragma unroll
  for (int it = 0; it < 2; ++it) {
    const int i = I0 + it * 16 + lo;
    const float cc = cw[i], ss = sw[i];
#pragma unroll
    for (int bt = 0; bt < 2; ++bt) {
#pragma unroll
      for (int r = 0; r < 8; ++r) {
        const float orr = aRR[it][bt][r] - aII[it][bt][r];   // Re(M x)
        const float oii = aRI[it][bt][r] + aIR[it][bt][r];   // Im(M x)
        const int b = B0 + bt * 16 + r + 8 * hf;
        outre[(size_t)b * HDIM + i] = orr * cc - oii * ss;
        outim[(size_t)b * HDIM + i] = orr * ss + oii * cc;
      }
    }
  }
}

// ---------------------------------------------------------------------------
extern "C" void kernel_launch(void* const* d_in, const int* in_sizes, int n_in,
                              void* d_out, int out_size, void* d_ws, size_t ws_size,
                              hipStream_t stream) {
  (void)in_sizes; (void)n_in; (void)out_size; (void)ws_size;
  const float* x_re       = (const float*)d_in[0];
  const float* x_im       = (const float*)d_in[1];
  const float* omega      = (const float*)d_in[2];
  const float* even_theta = (const float*)d_in[3];
  const float* odd_theta  = (const float*)d_in[4];
  const float* even_phi   = (const float*)d_in[5];
  const float* odd_phi    = (const float*)d_in[6];

  float* ws    = (float*)d_ws;
  float* MtrP  = ws;                                   // H*H f32 (4 MB)
  float* MtiP  = MtrP + (size_t)HDIM * HDIM;           // H*H f32 (4 MB)
  float* coeff = MtiP + (size_t)HDIM * HDIM;           // 64*2*4*H f32 (2 MB)
  float* cwv   = coeff + (size_t)NSTEP * 2 * 4 * HDIM; // H f32
  float* swv   = cwv + HDIM;                           // H f32

  eunn_precompute<<<(NSTEP * HDIM) / 256, 256, 0, stream>>>(
      omega, even_theta, odd_theta, even_phi, odd_phi, coeff, cwv, swv);

  eunn_build<<<HDIM / 8, 256, 0, stream>>>(coeff, MtrP, MtiP);

  // 32 i-groups x 8 blocks each; 8 waves/block, one 32x32 super-tile per wave
  eunn_gemm<<<256, 256, 0, stream>>>(x_re, x_im, MtrP, MtiP, cwv, swv,
                                     (float*)d_out);
}